// YoloOneLoss_18760417149025
// MI455X (gfx1250) — compile-verified
//
#include <hip/hip_runtime.h>
#include <hip/hip_bf16.h>

// ---------------------------------------------------------------------------
// YOLO loss for MI455X (gfx1250, wave32).
//
// Bandwidth plan: dense pass reads ONLY the objectness channel (1/5 of the
// det tensors, ~17.2 MB total -> ~0.75us at 23.3 TB/s).  Batch index is a
// grid dimension so the hot loop has no integer division.  Sparse pass
// handles the <=4096 target cells per scale with an atomicOr dedup bitmask
// in d_ws.  Wave reduction of partials uses V_WMMA_F32_16X16X4_F32.
// ---------------------------------------------------------------------------

typedef __attribute__((ext_vector_type(2))) float v2f;
typedef __attribute__((ext_vector_type(8))) float v8f;

#define FOCAL_A_C 0.25f
#define NEG_W_C   0.05f

// Focal loss at a cell with target t=0, logit x:
//   p = sigmoid(x); bce = softplus(x); floss = A * p^1.5 * bce
// Single-exp formulation: e = exp(-|x|);
//   sigmoid(x) = (x>=0) ? 1/(1+e) : e/(1+e);  softplus(x) = max(x,0)+log(1+e)
__device__ __forceinline__ float floss_neg(float x) {
    float e  = __expf(-fabsf(x));
    float r  = __builtin_amdgcn_rcpf(1.0f + e);
    float p  = (x >= 0.0f) ? r : e * r;               // sigmoid(x)
    float sp = fmaxf(x, 0.0f) + __logf(1.0f + e);     // softplus(x)
    return FOCAL_A_C * p * sqrtf(p) * sp;
}
// Focal loss with t=1:  A * (1-p)^1.5 * softplus(-x)  ==  floss_neg(-x)
__device__ __forceinline__ float floss_pos(float x) { return floss_neg(-x); }

__device__ __forceinline__ float fast_sigmoid(float x) {
    float e = __expf(-fabsf(x));
    float r = __builtin_amdgcn_rcpf(1.0f + e);
    return (x >= 0.0f) ? r : e * r;
}

// Full-wave (32 lane) sum on the matrix pipe.
// A (16x4): lane L<16 -> A[L,0]=s; lane L>=16 -> A[L-16,2]=s.  B = ones.
// D[m,n] = s(lane m) + s(lane m+16) for every n.  Lane<16 VGPR v holds
// D[v,lane]; lane>=16 holds D[8+v,lane-16].  So the 8-reg sum at lane0 is
// rows 0..7 and at lane16 rows 8..15; broadcast-add gives the wave total.
__device__ __forceinline__ float wave_sum_wmma(float s) {
    v2f a; a[0] = s;    a[1] = 0.0f;
    v2f b; b[0] = 1.0f; b[1] = 1.0f;
    v8f c = {};
    c = __builtin_amdgcn_wmma_f32_16x16x4_f32(
        /*neg_a=*/false, a, /*neg_b=*/false, b,
        /*c_mod=*/(short)0, c, /*reuse_a=*/false, /*reuse_b=*/false);
    float loc = ((c[0] + c[1]) + (c[2] + c[3])) + ((c[4] + c[5]) + (c[6] + c[7]));
    return __shfl(loc, 0, 32) + __shfl(loc, 16, 32);
}

// ---------------------------------------------------------------------------
// Workspace layout (32-bit words):
//   [0..31]   per-scale accumulators: slot s*8 + {0:neg_all(f32), 1:neg_corr(f32),
//                                                 2:pos_sum(f32), 3:box_sum(f32),
//                                                 4:n(u32)}
//   [32.. ]   dedup bitmasks: p3 (102400 w), p4 (25600 w), p5 (6400 w)
// ---------------------------------------------------------------------------

__global__ void zero_ws_kernel(unsigned* __restrict__ w, long n) {
    long i = (long)blockIdx.x * blockDim.x + threadIdx.x;
    long stride = (long)gridDim.x * blockDim.x;
    for (; i < n; i += stride) w[i] = 0u;
}

// Dense pass: stream objectness plane (channel 4) of one head, sum floss(t=0).
// blockIdx.y = batch index -> no integer division in the hot loop.
__global__ void neg_obj_kernel(const float* __restrict__ det, int HW,
                               float* __restrict__ acc_neg_all) {
    const int hw4 = HW >> 2;                  // float4 per plane (HW % 4 == 0)
    // channel-4 plane base: (b*5 + 4) * HW floats -> 16B aligned
    const float4* plane =
        (const float4*)(det + ((long)blockIdx.y * 5 + 4) * (long)HW);
    const int stride = gridDim.x * blockDim.x;
    float s = 0.0f;
    for (int o = blockIdx.x * blockDim.x + threadIdx.x; o < hw4; o += stride) {
        float4 v = plane[o];
        s += floss_neg(v.x) + floss_neg(v.y) + floss_neg(v.z) + floss_neg(v.w);
    }
    // all 256 threads reconverge here -> EXEC all-1s for WMMA
    float wsum = wave_sum_wmma(s);
    __shared__ float smem[8];
    int lane = threadIdx.x & 31, wv = threadIdx.x >> 5;
    if (lane == 0) smem[wv] = wsum;
    __syncthreads();
    if (threadIdx.x == 0) {
        float t = 0.0f;
        int nw = blockDim.x >> 5;
        for (int k = 0; k < nw; ++k) t += smem[k];
        atomicAdd(acc_neg_all, t);
    }
}

// Sparse pass: one thread per (scale, target). Dedup via bitmask; the winner
// gathers the 5 channels at its cell and accumulates box / pos / neg-correction.
__global__ void sparse_targets_kernel(const float* __restrict__ tg, int N,
                                      const float* __restrict__ d0,
                                      const float* __restrict__ d1,
                                      const float* __restrict__ d2,
                                      unsigned* __restrict__ m0,
                                      unsigned* __restrict__ m1,
                                      unsigned* __restrict__ m2,
                                      float* __restrict__ accf,
                                      unsigned* __restrict__ accu) {
    int tid = blockIdx.x * blockDim.x + threadIdx.x;
    if (tid >= 3 * N) return;
    int s = tid / N;
    int i = tid - s * N;

    const float* det; unsigned* mask; int H, W;
    if (s == 0)      { det = d0; mask = m0; H = 160; W = 160; }
    else if (s == 1) { det = d1; mask = m1; H = 80;  W = 80;  }
    else             { det = d2; mask = m2; H = 40;  W = 40;  }

    const float* tr = tg + (long)i * 6;
    int   b  = (int)tr[0];
    float x  = tr[2], y = tr[3], w = tr[4], h = tr[5];
    float gx = x * (float)W, gy = y * (float)H;
    int gi = (int)fminf(fmaxf(gx, 0.0f), (float)(W - 1));
    int gj = (int)fminf(fmaxf(gy, 0.0f), (float)(H - 1));

    long cid = ((long)b * H + gj) * W + gi;
    unsigned word = (unsigned)(cid >> 5);
    unsigned bit  = (unsigned)cid & 31u;
    unsigned old  = atomicOr(&mask[word], 1u << bit);
    if ((old >> bit) & 1u) return;          // another target already owns this cell

    long HW   = (long)H * W;
    long base = (long)b * 5 * HW + (long)gj * W + gi;
    float p0 = det[base + 0 * HW];
    float p1 = det[base + 1 * HW];
    float p2 = det[base + 2 * HW];
    float p3 = det[base + 3 * HW];
    float p4 = det[base + 4 * HW];

    float vx = gx - (float)gi, vy = gy - (float)gj;
    float vw = w * (float)W,   vh = h * (float)H;
    float sx = fast_sigmoid(p0), sy = fast_sigmoid(p1);
    float box = (sx - vx) * (sx - vx) + (sy - vy) * (sy - vy)
              + (p2 - vw) * (p2 - vw) + (p3 - vh) * (p3 - vh);

    atomicAdd(&accf[s * 8 + 1], floss_neg(p4));  // subtract from dense neg sum
    atomicAdd(&accf[s * 8 + 2], floss_pos(p4));  // pos focal term
    atomicAdd(&accf[s * 8 + 3], box);            // box MSE sum
    atomicAdd(&accu[s * 8 + 4], 1u);             // n (unique cells)
}

__global__ void finalize_kernel(const float* __restrict__ accf,
                                const unsigned* __restrict__ accu,
                                float* __restrict__ out) {
    if (threadIdx.x != 0 || blockIdx.x != 0) return;
    const float bw[3]    = {7.5f, 7.5f, 7.5f * 1.2f};
    const float cells[3] = {128.0f * 160.0f * 160.0f,
                            128.0f * 80.0f * 80.0f,
                            128.0f * 40.0f * 40.0f};
    float total = 0.0f;
    for (int s = 0; s < 3; ++s) {
        float n  = (float)accu[s * 8 + 4];
        float na = accf[s * 8 + 0];   // sum floss(t=0) over ALL cells
        float nc = accf[s * 8 + 1];   // sum floss(t=0) over masked cells
        float ps = accf[s * 8 + 2];   // sum floss(t=1) over masked cells
        float bs = accf[s * 8 + 3];   // box MSE sum over masked cells
        float nd = fmaxf(n, 1.0f);
        float box_loss = (n > 0.0f) ? bs / (nd * 2.0f) : 0.0f;
        float pos      = (n > 0.0f) ? ps / nd : 0.0f;
        float nneg     = cells[s] - n;
        float neg      = (nneg > 0.0f) ? (na - nc) / nneg : 0.0f;
        total += box_loss * bw[s] + (pos + NEG_W_C * neg);  // OBJ_W = 1.0
    }
    out[0] = total;
}

extern "C" void kernel_launch(void* const* d_in, const int* in_sizes, int n_in,
                              void* d_out, int out_size, void* d_ws, size_t ws_size,
                              hipStream_t stream) {
    const float* p3 = (const float*)d_in[0];   // [128,5,160,160]
    const float* p4 = (const float*)d_in[1];   // [128,5,80,80]
    const float* p5 = (const float*)d_in[2];   // [128,5,40,40]
    const float* tg = (const float*)d_in[3];   // [N,6]
    float*    out  = (float*)d_out;
    unsigned* wsu  = (unsigned*)d_ws;
    float*    wsf  = (float*)d_ws;

    const int  B  = 128;
    const long accWords = 32;
    const long w0 = 128L * 160 * 160 / 32;     // 102400 mask words (p3)
    const long w1 = 128L * 80 * 80 / 32;       //  25600 mask words (p4)
    const long w2 = 128L * 40 * 40 / 32;       //   6400 mask words (p5)
    const long m0 = accWords;
    const long m1 = m0 + w0;
    const long m2 = m1 + w1;
    const long totalWords = m2 + w2;           // ~537 KB of workspace

    // 1) zero accumulators + dedup bitmasks (must be re-done every call)
    zero_ws_kernel<<<256, 256, 0, stream>>>(wsu, totalWords);

    // 2) dense objectness-only focal sums (channel 4 only -> 1/5 the bytes)
    //    grid.x chosen so each thread runs <=2 float4 iterations:
    //    p3: hw4=6400 -> 13 blocks/plane; p4: hw4=1600 -> 4; p5: hw4=400 -> 1
    neg_obj_kernel<<<dim3(13, B), 256, 0, stream>>>(p3, 160 * 160, &wsf[0 * 8 + 0]);
    neg_obj_kernel<<<dim3(4,  B), 256, 0, stream>>>(p4, 80 * 80,   &wsf[1 * 8 + 0]);
    neg_obj_kernel<<<dim3(1,  B), 256, 0, stream>>>(p5, 40 * 40,   &wsf[2 * 8 + 0]);

    // 3) sparse per-target pass (dedup + box + pos + neg correction)
    int N = in_sizes[3] / 6;
    int nthreads = 3 * N;
    sparse_targets_kernel<<<(nthreads + 255) / 256, 256, 0, stream>>>(
        tg, N, p3, p4, p5, wsu + m0, wsu + m1, wsu + m2, wsf, wsu);

    // 4) combine
    finalize_kernel<<<1, 32, 0, stream>>>(wsf, wsu, out);
}